// ConvAttention_34694745817831
// MI455X (gfx1250) — compile-verified
//
#include <hip/hip_runtime.h>
#include <cstddef>

// CDNA5 fp32 WMMA: D(16x16,f32) = A(16x4,f32) * B(4x16,f32) + C
typedef float v2f __attribute__((ext_vector_type(2)));
typedef float v8f __attribute__((ext_vector_type(8)));

__device__ __forceinline__ v8f wmma_f32(v2f a, v2f b, v8f c) {
  // 8 args: (neg_a, A, neg_b, B, c_mod, C, reuse_a, reuse_b)
  return __builtin_amdgcn_wmma_f32_16x16x4_f32(false, a, false, b, (short)0, c,
                                               false, false);
}

// Problem constants
#define CDIM 128   // channels
#define HD   64    // q/k head dim
#define SEQ  512   // tokens per (b,h)
#define NBH  512   // 8*64 batch-head pairs

// ---------------------------------------------------------------------------
// Async global->LDS staging (CDNA5 GLOBAL_LOAD_ASYNC_TO_LDS_B128, ASYNCcnt).
// Falls back to synchronous float4 copies if the builtin is unavailable.
// Builtin signature (per hipcc diagnostic): (int4* src_global, int4* dst_lds,
// imm offset, imm cpol) with generic pointers.
// ---------------------------------------------------------------------------
#if defined(__HIP_DEVICE_COMPILE__) && \
    __has_builtin(__builtin_amdgcn_global_load_async_to_lds_b128)
#define ASYNC_LDS 1
typedef int v4i __attribute__((vector_size(16)));
#else
#define ASYNC_LDS 0
#endif

__device__ __forceinline__ void cp_async_f4(float* dstLds, const float* srcGlb,
                                            int nf4, int tid) {
#if ASYNC_LDS
  for (int i = tid; i < nf4; i += 512)
    __builtin_amdgcn_global_load_async_to_lds_b128(
        (v4i*)(srcGlb + 4 * i), (v4i*)(dstLds + 4 * i), 0, 0);
#else
  for (int i = tid; i < nf4; i += 512)
    ((float4*)dstLds)[i] = ((const float4*)srcGlb)[i];
#endif
}

__device__ __forceinline__ void cp_async_wait() {
#if ASYNC_LDS
#if __has_builtin(__builtin_amdgcn_s_wait_asynccnt)
  __builtin_amdgcn_s_wait_asynccnt(0);
#else
  asm volatile("s_wait_asynccnt 0x0" ::: "memory");
#endif
#endif
}

// ---------------------------------------------------------------------------
// Kernel 1: per-bh block. Project K (all 512 keys) into LDS, then for each
// 32-row query chunk: project Q, scores = Q*K^T via WMMA, row softmax in LDS,
// write final softmax weights w to global.
// ---------------------------------------------------------------------------
__global__ void __launch_bounds__(512)
qk_softmax_kernel(const float* __restrict__ x, const float* __restrict__ Wqkv,
                  float* __restrict__ wOut)
{
  __shared__ float sW[CDIM * HD];    // 32 KB  (Wk, later Wq; ld=64)
  __shared__ float sK[SEQ * HD];     // 128 KB (all projected keys; ld=64)
  __shared__ float sX0[64 * CDIM];   // 32 KB  (x staging, double buffered)
  __shared__ float sX1[64 * CDIM];   // 32 KB
  __shared__ float sQ[32 * HD];      // 8 KB   (q chunk; ld=64)
  __shared__ float sS[32 * SEQ];     // 64 KB  (score rows; ld=512)
  __shared__ float sRed[32 * 16];    // 2 KB   (softmax partials)

  const int bh   = blockIdx.x;
  const int tid  = threadIdx.x;
  const int wv   = tid >> 5;   // wave id 0..15
  const int lane = tid & 31;
  const int m    = lane & 15;  // row (A) / col (B,C,D) within tile
  const int kh   = lane >> 4;  // K-half select

  const float* xb = x + (size_t)bh * (SEQ * CDIM);
  float*       wb = wOut + (size_t)bh * (SEQ * SEQ);

  // Kick off first x chunk (keys 0..63) asynchronously.
  cp_async_f4(sX0, xb, 2048, tid);

  // Stage Wk = W_qkv[:, 64:128]  (row-major, ld=64)
  for (int i = tid; i < CDIM * HD; i += 512)
    sW[i] = Wqkv[(i >> 6) * 256 + 64 + (i & 63)];

  // ---- K projection: 8 chunks of 64 keys, 16 tiles per chunk (one per wave)
  for (int c = 0; c < 8; ++c) {
    float* cur = (c & 1) ? sX1 : sX0;
    cp_async_wait();
    __syncthreads();
    if (c < 7)  // prefetch next chunk into the other buffer
      cp_async_f4((c & 1) ? sX0 : sX1, xb + (c + 1) * (64 * CDIM), 2048, tid);

    const int mt = wv >> 2, nt = wv & 3;   // 4x4 tiles over 64x64
    v8f acc = {};
#pragma unroll
    for (int ks = 0; ks < 32; ++ks) {      // K = 128
      const int k0 = ks * 4 + 2 * kh;
      v2f a, b;
      a.x = cur[(mt * 16 + m) * CDIM + k0];
      a.y = cur[(mt * 16 + m) * CDIM + k0 + 1];
      b.x = sW[k0 * HD + nt * 16 + m];
      b.y = sW[(k0 + 1) * HD + nt * 16 + m];
      acc = wmma_f32(a, b, acc);
    }
#pragma unroll
    for (int r = 0; r < 8; ++r)
      sK[(c * 64 + mt * 16 + 8 * kh + r) * HD + nt * 16 + m] = acc[r];
  }
  __syncthreads();
  // Stage Wq = W_qkv[:, 0:64]; kick off first query-chunk x load.
  cp_async_f4(sX0, xb, 1024, tid);
  for (int i = tid; i < CDIM * HD; i += 512)
    sW[i] = Wqkv[(i >> 6) * 256 + (i & 63)];

  // ---- Per 32-row query chunk: Q proj -> scores -> softmax -> store w
  for (int qc = 0; qc < 16; ++qc) {
    float* cur = (qc & 1) ? sX1 : sX0;
    cp_async_wait();
    __syncthreads();
    if (qc < 15)
      cp_async_f4((qc & 1) ? sX0 : sX1, xb + (qc + 1) * (32 * CDIM), 1024, tid);

    if (wv < 8) {                          // Q proj: 2x4 tiles over 32x64
      const int mt = wv >> 2, nt = wv & 3;
      v8f acc = {};
#pragma unroll
      for (int ks = 0; ks < 32; ++ks) {    // K = 128
        const int k0 = ks * 4 + 2 * kh;
        v2f a, b;
        a.x = cur[(mt * 16 + m) * CDIM + k0];
        a.y = cur[(mt * 16 + m) * CDIM + k0 + 1];
        b.x = sW[k0 * HD + nt * 16 + m];
        b.y = sW[(k0 + 1) * HD + nt * 16 + m];
        acc = wmma_f32(a, b, acc);
      }
#pragma unroll
      for (int r = 0; r < 8; ++r)
        sQ[(mt * 16 + 8 * kh + r) * HD + nt * 16 + m] = acc[r];
    }
    __syncthreads();

    {  // scores (32 x 512) = Q * K^T ; wave: 2 M-tiles x 2 N-tiles (B reused)
      const int n0 = wv * 32;
      v8f acc[2][2] = {};
#pragma unroll
      for (int ks = 0; ks < 16; ++ks) {    // K = 64 (head dim)
        const int k0 = ks * 4 + 2 * kh;
        v2f a0, a1;
        a0.x = sQ[m * HD + k0];
        a0.y = sQ[m * HD + k0 + 1];
        a1.x = sQ[(16 + m) * HD + k0];
        a1.y = sQ[(16 + m) * HD + k0 + 1];
#pragma unroll
        for (int nt = 0; nt < 2; ++nt) {
          v2f b;  // B[k][n] = K[n][k]
          b.x = sK[(n0 + nt * 16 + m) * HD + k0];
          b.y = sK[(n0 + nt * 16 + m) * HD + k0 + 1];
          acc[0][nt] = wmma_f32(a0, b, acc[0][nt]);
          acc[1][nt] = wmma_f32(a1, b, acc[1][nt]);
        }
      }
#pragma unroll
      for (int mt = 0; mt < 2; ++mt)
#pragma unroll
        for (int nt = 0; nt < 2; ++nt)
#pragma unroll
          for (int r = 0; r < 8; ++r)
            sS[(mt * 16 + 8 * kh + r) * SEQ + n0 + nt * 16 + m] = acc[mt][nt][r];
    }
    __syncthreads();

    {  // softmax: 16 threads per row, 32 elems each
      const int row = tid >> 4;
      const int seg = (tid & 15) * 32;
      float* srow = sS + row * SEQ + seg;
      float mx = -3.0e38f;
#pragma unroll 8
      for (int j = 0; j < 32; ++j) mx = fmaxf(mx, srow[j]);
      sRed[row * 16 + (tid & 15)] = mx;
      __syncthreads();
      float rmx = sRed[row * 16];
#pragma unroll
      for (int j = 1; j < 16; ++j) rmx = fmaxf(rmx, sRed[row * 16 + j]);
      __syncthreads();
      float sum = 0.f;
#pragma unroll 8
      for (int j = 0; j < 32; ++j) {
        const float e = __expf(srow[j] - rmx);
        srow[j] = e;
        sum += e;
      }
      sRed[row * 16 + (tid & 15)] = sum;
      __syncthreads();
      float rs = 0.f;
#pragma unroll
      for (int j = 0; j < 16; ++j) rs += sRed[row * 16 + j];
      const float inv = 1.0f / rs;
      float* dst = wb + (size_t)(qc * 32 + row) * SEQ + seg;
#pragma unroll
      for (int j = 0; j < 32; j += 4) {
        float4 v;
        v.x = srow[j + 0] * inv; v.y = srow[j + 1] * inv;
        v.z = srow[j + 2] * inv; v.w = srow[j + 3] * inv;
        *(float4*)(dst + j) = v;
      }
    }
  }
}

// ---------------------------------------------------------------------------
// Kernel 2: per (bh, 256-row half). Re-project V chunk-by-chunk into LDS,
// accumulate y = w * V with WMMA (A from global w, B from LDS), add residual.
// ---------------------------------------------------------------------------
__global__ void __launch_bounds__(512)
pv_residual_kernel(const float* __restrict__ x, const float* __restrict__ Wqkv,
                   const float* __restrict__ wIn, float* __restrict__ yOut)
{
  __shared__ float sW[CDIM * CDIM];  // 64 KB (Wv; ld=128)
  __shared__ float sX0[64 * CDIM];   // 32 KB (x staging, double buffered)
  __shared__ float sX1[64 * CDIM];   // 32 KB
  __shared__ float sV[64 * CDIM];    // 32 KB (projected v chunk)

  const int bh   = blockIdx.x >> 1;
  const int half = blockIdx.x & 1;
  const int tid  = threadIdx.x;
  const int wv   = tid >> 5;
  const int lane = tid & 31;
  const int m    = lane & 15;
  const int kh   = lane >> 4;

  const float* xb = x + (size_t)bh * (SEQ * CDIM);
  const float* wb = wIn + (size_t)bh * (SEQ * SEQ) + (size_t)half * 256 * SEQ;
  float*       yb = yOut + (size_t)bh * (SEQ * CDIM);

  cp_async_f4(sX0, xb, 2048, tid);

  // Stage Wv = W_qkv[:, 128:256]  (ld=128)
  for (int i = tid; i < CDIM * CDIM / 4; i += 512) {
    const int r = i >> 5, c4 = (i & 31) * 4;
    *(float4*)(sW + r * CDIM + c4) = *(const float4*)(Wqkv + r * 256 + 128 + c4);
  }

  const int rowG = wv >> 1;          // 0..7  -> 32 query rows each
  const int colG = wv & 1;           // 0..1  -> 64 v-cols each
  const int row0 = rowG * 32;
  const int col0 = colG * 64;

  v8f acc[2][4] = {};                // 32 rows x 64 cols per wave

  for (int kc = 0; kc < 8; ++kc) {   // 64 keys per chunk
    float* cur = (kc & 1) ? sX1 : sX0;
    cp_async_wait();
    __syncthreads();
    if (kc < 7)
      cp_async_f4((kc & 1) ? sX0 : sX1, xb + (kc + 1) * (64 * CDIM), 2048, tid);

    // V projection (64 x 128): wave = 2 M-tiles x 1 N-tile (B fragment reused)
    {
      const int nt  = wv & 7;
      const int mt0 = (wv >> 3) * 2;
      v8f va0 = {}, va1 = {};
#pragma unroll
      for (int ks = 0; ks < 32; ++ks) {  // K = 128
        const int k0 = ks * 4 + 2 * kh;
        v2f a0, a1, b;
        b.x = sW[k0 * CDIM + nt * 16 + m];
        b.y = sW[(k0 + 1) * CDIM + nt * 16 + m];
        a0.x = cur[(mt0 * 16 + m) * CDIM + k0];
        a0.y = cur[(mt0 * 16 + m) * CDIM + k0 + 1];
        a1.x = cur[((mt0 + 1) * 16 + m) * CDIM + k0];
        a1.y = cur[((mt0 + 1) * 16 + m) * CDIM + k0 + 1];
        va0 = wmma_f32(a0, b, va0);
        va1 = wmma_f32(a1, b, va1);
      }
#pragma unroll
      for (int r = 0; r < 8; ++r) {
        sV[(mt0 * 16 + 8 * kh + r) * CDIM + nt * 16 + m]       = va0[r];
        sV[((mt0 + 1) * 16 + 8 * kh + r) * CDIM + nt * 16 + m] = va1[r];
      }
    }
    __syncthreads();

    // y += w[:, kc*64 : kc*64+64] * V_chunk
#pragma unroll
    for (int ks = 0; ks < 16; ++ks) {    // K = 64 keys
      const int k0 = ks * 4 + 2 * kh;
      const float* wr0 = wb + (size_t)(row0 + m) * SEQ + kc * 64 + k0;
      const float* wr1 = wb + (size_t)(row0 + 16 + m) * SEQ + kc * 64 + k0;
      v2f a0, a1;
      a0.x = wr0[0]; a0.y = wr0[1];
      a1.x = wr1[0]; a1.y = wr1[1];
#pragma unroll
      for (int nt = 0; nt < 4; ++nt) {
        v2f b;
        b.x = sV[k0 * CDIM + col0 + nt * 16 + m];
        b.y = sV[(k0 + 1) * CDIM + col0 + nt * 16 + m];
        acc[0][nt] = wmma_f32(a0, b, acc[0][nt]);
        acc[1][nt] = wmma_f32(a1, b, acc[1][nt]);
      }
    }
    __syncthreads();
  }

  // residual + store y
#pragma unroll
  for (int mt2 = 0; mt2 < 2; ++mt2)
#pragma unroll
    for (int nt = 0; nt < 4; ++nt)
#pragma unroll
      for (int r = 0; r < 8; ++r) {
        const int grow = half * 256 + row0 + mt2 * 16 + 8 * kh + r;
        const int gcol = col0 + nt * 16 + m;
        yb[(size_t)grow * CDIM + gcol] =
            acc[mt2][nt][r] + xb[(size_t)grow * CDIM + gcol];
      }
}

extern "C" void kernel_launch(void* const* d_in, const int* in_sizes, int n_in,
                              void* d_out, int out_size, void* d_ws, size_t ws_size,
                              hipStream_t stream) {
  (void)in_sizes; (void)n_in; (void)d_ws; (void)ws_size; (void)out_size;
  const float* x = (const float*)d_in[0];
  const float* W = (const float*)d_in[1];
  float* y = (float*)d_out;                                  // 8*64*512*128
  float* w = (float*)d_out + (size_t)8 * 64 * 512 * CDIM;    // 8*64*512*512

  qk_softmax_kernel<<<NBH, 512, 0, stream>>>(x, W, w);
  pv_residual_kernel<<<NBH * 2, 512, 0, stream>>>(x, W, w, y);
}